// FFM_model_30408368456214
// MI455X (gfx1250) — compile-verified
//
#include <hip/hip_runtime.h>
#include <math.h>

// FFM scoring kernel for MI455X (gfx1250, wave32).
// Memory-bound streaming reduction (~509 MB in, 64 KB out -> ~22 us floor at
// 23.3 TB/s). Row/column sums run on the matrix pipe with
// V_WMMA_F32_16X16X4_F32 using an all-ones A matrix (D[m][n] = sum_k B[k][n]),
// which is invariant to B's K-slot lane layout; VALU co-executes the
// sum-of-squares FMAs. One wave per batch element. All OOB masking is folded
// at compile time (R mod 4 in {0,2}); loads are non-temporal (single-pass
// stream, zero reuse).

typedef __attribute__((ext_vector_type(2))) float v2f;
typedef __attribute__((ext_vector_type(8))) float v8f;

static constexpr int W  = 97;   // 3*32 + 1 (last col = linear/bias)
static constexpr int RU = 50;
static constexpr int RC = 10;
static constexpr int RD = 20;
static constexpr int WAVES_PER_BLOCK = 8;

// Reduce one field matrix X[R x 97] for this batch element:
//  s[t]  (t=0..5): column sums of 16-col tile t, held per-lane at col = 16t + (lane&15)
//                  (duplicated across the two lane halves -> counted 2x in lane sums)
//  sq    : per-lane partial sum of squares over tiles SQT, SQT+1
//  bias  : per-lane partial sum of column 96
template <int R, int SQT>
__device__ __forceinline__ void field_reduce(const float* __restrict__ X,
                                             float s[6], float& sq, float& bias) {
    static_assert(R % 4 == 0 || R % 4 == 2, "tail folding assumes R mod 4 in {0,2}");

    const int lane = threadIdx.x & 31;
    const int half = lane >> 4;    // which of the two rows this lane supplies per B VGPR
    const int colb = lane & 15;    // column within 16-wide tile (WMMA N = lane%16)

    const v2f a = {1.0f, 1.0f};    // all-ones A matrix -> WMMA performs K-dim column sum

    v8f acc[6];
#pragma unroll
    for (int t = 0; t < 6; ++t) acc[t] = v8f{};

    // One per-lane base; every access below is base + compile-time 24-bit offset.
    const float* baseA = X + half * W + colb;        // rows r0 + half      (always valid)
    const float* baseB = X + (half + 2) * W + colb;  // rows r0 + 2 + half  (valid iff r0+2 < R)

    float sql = 0.0f;

#pragma unroll
    for (int r0 = 0; r0 < R; r0 += 4) {              // 4 rows (K=4) per WMMA
        const bool okB = (r0 + 2 < R);               // compile-time after unroll
#pragma unroll
        for (int t = 0; t < 6; ++t) {                // cols 0..95 contiguous per chunk
            const int off = r0 * W + t * 16;
            v2f b;
            b.x = __builtin_nontemporal_load(baseA + off);
            b.y = okB ? __builtin_nontemporal_load(baseB + off) : 0.0f;
            if (t == SQT || t == SQT + 1)            // this field's own 32-col FM block
                sql = fmaf(b.x, b.x, fmaf(b.y, b.y, sql));
            acc[t] = __builtin_amdgcn_wmma_f32_16x16x4_f32(
                false, a, false, b, (short)0, acc[t], false, false);
        }
    }

#pragma unroll
    for (int t = 0; t < 6; ++t) s[t] = acc[t][0];    // D vgpr0 = colsum at N = lane%16

    // Bias column (col 96), strided across lanes; outside the hot loop so its
    // loadcnt wait cannot drain the tile-load pipeline.
    float bl = 0.0f;
#pragma unroll
    for (int r = lane; r < R; r += 32)
        bl += __builtin_nontemporal_load(X + r * W + 96);

    sq   += sql;
    bias += bl;
}

__global__ void __launch_bounds__(WAVES_PER_BLOCK * 32)
ffm_score_kernel(const float* __restrict__ user,
                 const float* __restrict__ ctx,
                 const float* __restrict__ doc,
                 float* __restrict__ out, int B) {
    const int wave = blockIdx.x * (blockDim.x >> 5) + (threadIdx.x >> 5);
    if (wave >= B) return;

    const float* ub = user + (size_t)wave * RU * W;
    const float* cb = ctx  + (size_t)wave * RC * W;
    const float* db = doc  + (size_t)wave * RD * W;

    float su[6], sc[6], sd[6];
    float squ = 0.0f, sqc = 0.0f, sqd = 0.0f, bsum = 0.0f;

    field_reduce<RU, 0>(ub, su, squ, bsum);  // user FM block: cols  0..31 (tiles 0,1)
    field_reduce<RC, 2>(cb, sc, sqc, bsum);  // ctx  FM block: cols 32..63 (tiles 2,3)
    field_reduce<RD, 4>(db, sd, sqd, bsum);  // doc  FM block: cols 64..95 (tiles 4,5)

    // Per-lane partial. s[] values are duplicated in both lane halves (x2 in the
    // lane sum), sq/bias partials are counted once. FM term carries 0.5.
    float p = 0.25f * (su[0] * su[0] + su[1] * su[1]      // sum(su[0:32]^2)
                     + sc[2] * sc[2] + sc[3] * sc[3]      // sum(sc[32:64]^2)
                     + sd[4] * sd[4] + sd[5] * sd[5])     // sum(sd[64:96]^2)
            - 0.5f  * (squ + sqc + sqd)                   // -0.5 * sum of squares
            + 0.5f  * (su[2] * sc[0] + su[3] * sc[1]      // su[32:64] . sc[0:32]
                     + su[4] * sd[0] + su[5] * sd[1]      // su[64:96] . sd[0:32]
                     + sc[4] * sd[2] + sc[5] * sd[3])     // sc[64:96] . sd[32:64]
            + bsum;                                       // linear/bias column

#pragma unroll
    for (int off = 16; off > 0; off >>= 1)                // wave32 butterfly reduce
        p += __shfl_xor(p, off, 32);

    if ((threadIdx.x & 31) == 0)
        out[wave] = 1.0f / (1.0f + expf(-p));
}

extern "C" void kernel_launch(void* const* d_in, const int* in_sizes, int n_in,
                              void* d_out, int out_size, void* d_ws, size_t ws_size,
                              hipStream_t stream) {
    const float* user = (const float*)d_in[0];
    const float* ctx  = (const float*)d_in[1];
    const float* doc  = (const float*)d_in[2];
    float*       out  = (float*)d_out;

    const int B = in_sizes[0] / (RU * W);

    const int threads = WAVES_PER_BLOCK * 32;
    const int blocks  = (B + WAVES_PER_BLOCK - 1) / WAVES_PER_BLOCK;
    ffm_score_kernel<<<blocks, threads, 0, stream>>>(user, ctx, doc, out, B);
}